// RAiDAttentiveWalk_50783693308065
// MI455X (gfx1250) — compile-verified
//
#include <hip/hip_runtime.h>
#include <hip/hip_bf16.h>

typedef __attribute__((ext_vector_type(2))) float v2f;
typedef __attribute__((ext_vector_type(8))) float v8f;

#define EMB      128
#define WALK     40
#define PADM     48
#define LSTRIDE  130   // LDS row stride (floats): even (8B-aligned b64 loads), 130%64=2 -> conflict-free column reads
#define GSTRIDE  50
#define NPAIR    370
#define BATCH    1024
#define TOTAL_PAIRS (BATCH * NPAIR)      // 378880
#define NEG_BLOCKS  2048
#define POS_THREADS 288                  // 9 waves: one per 16x16 tile of the 48x48 Gram matrix

__device__ __forceinline__ float softplus_f(float y) {
    // log(1 + exp(y)), numerically stable
    return fmaxf(y, 0.f) + log1pf(expf(-fabsf(y)));
}

// ---------------------------------------------------------------------------
// Reconstruct the (src,dst) window-pair tables for L=40, W=5 (370 pairs).
// ---------------------------------------------------------------------------
__global__ void build_pairs_kernel(int* __restrict__ idx_src, int* __restrict__ idx_dst) {
    int p = threadIdx.x;
    if (p >= NPAIR) return;
    int c = 0;
    for (int i = 0; i < WALK; ++i) {
        int lo = i - 5; if (lo < 0) lo = 0;
        int cntL = i - lo;
        int hi = i + 6; if (hi > WALK) hi = WALK;
        int cntR = hi - i - 1;
        int tot = cntL + cntR;
        if (p < c + tot) {
            int r = p - c;
            idx_dst[p] = i;
            idx_src[p] = (r < cntL) ? (lo + r) : (i + 1 + (r - cntL));
            return;
        }
        c += tot;
    }
}

// ---------------------------------------------------------------------------
// Positive pairs: per-walk 48x48x128 Gram matrix via V_WMMA_F32_16X16X4_F32.
// ---------------------------------------------------------------------------
__global__ __launch_bounds__(POS_THREADS) void pos_kernel(
    const float* __restrict__ node_embed,
    const float* __restrict__ ctx_embed,
    const int*   __restrict__ batch_walk,
    const int*   __restrict__ idx_src,
    const int*   __restrict__ idx_dst,
    double*      __restrict__ partial)
{
    __shared__ int   walkid[WALK];
    __shared__ float Nl[PADM * LSTRIDE];
    __shared__ float Cl[PADM * LSTRIDE];
    __shared__ float Gl[PADM * GSTRIDE];
    __shared__ float red[12];

    const int tid = threadIdx.x;
    const int b   = blockIdx.x;

    if (tid < WALK) walkid[tid] = batch_walk[b * WALK + tid];
    __syncthreads();

    // Stage the walk's node/context rows into LDS (rows 40..47 zero-padded).
    for (int idx = tid; idx < PADM * EMB; idx += POS_THREADS) {
        int row = idx >> 7, col = idx & 127;
        float nv = 0.f, cv = 0.f;
        if (row < WALK) {
            long long base = (long long)walkid[row] * EMB + col;
            nv = node_embed[base];
            cv = ctx_embed[base];
        }
        Nl[row * LSTRIDE + col] = nv;
        Cl[row * LSTRIDE + col] = cv;
    }
    __syncthreads();

    // 9 waves, one 16x16 output tile each: G = N (48x128) * C^T (128x48).
    const int w    = tid >> 5;
    const int lane = tid & 31;
    const int M0   = (w / 3) * 16;
    const int N0   = (w % 3) * 16;
    const int rlo   = lane & 15;          // M (A) / N (B) within tile
    const int khalf = (lane >> 4) * 2;    // K sub-offset per half-wave

    const float* arow = &Nl[(M0 + rlo) * LSTRIDE + khalf];
    const float* brow = &Cl[(N0 + rlo) * LSTRIDE + khalf];

    v8f acc = {0.f, 0.f, 0.f, 0.f, 0.f, 0.f, 0.f, 0.f};
    #pragma unroll
    for (int kb = 0; kb < EMB; kb += 4) {
        v2f af = *(const v2f*)(arow + kb);   // A: lane l, reg v -> (M=l%16, K=kb+(l/16)*2+v)
        v2f bf = *(const v2f*)(brow + kb);   // B: lane l, reg v -> (N=l%16, K=kb+(l/16)*2+v)
        acc = __builtin_amdgcn_wmma_f32_16x16x4_f32(
            /*neg_a=*/false, af, /*neg_b=*/false, bf,
            /*c_mod=*/(short)0, acc, /*reuse_a=*/false, /*reuse_b=*/false);
    }

    // D layout: reg r, lane l -> (M = r + (l/16)*8, N = l%16)
    #pragma unroll
    for (int r = 0; r < 8; ++r) {
        Gl[(M0 + r + (lane >> 4) * 8) * GSTRIDE + N0 + rlo] = acc[r];
    }
    __syncthreads();

    // Gather the 370 scored entries, clip, softplus(-x).
    float local = 0.f;
    for (int p = tid; p < NPAIR; p += POS_THREADS) {
        float x = Gl[idx_src[p] * GSTRIDE + idx_dst[p]];
        x = fminf(6.f, fmaxf(-6.f, x));
        local += softplus_f(-x);
    }
    for (int off = 16; off; off >>= 1) local += __shfl_xor(local, off, 32);
    if (lane == 0) red[w] = local;
    __syncthreads();
    if (tid == 0) {
        float s = 0.f;
        #pragma unroll
        for (int i = 0; i < 9; ++i) s += red[i];
        partial[b] = (double)s;
    }
}

// ---------------------------------------------------------------------------
// Negative samples: one 128-d dot per wave; 5 negs share one A-row load.
// ---------------------------------------------------------------------------
__global__ __launch_bounds__(256) void neg_kernel(
    const float* __restrict__ node_embed,
    const float* __restrict__ ctx_embed,
    const int*   __restrict__ batch_walk,     // flat [BATCH*WALK]
    const int*   __restrict__ neg_dst_idx,    // [TOTAL_PAIRS*5] in [0, BATCH*WALK)
    const int*   __restrict__ idx_dst,
    double*      __restrict__ partial)
{
    __shared__ float red[8];
    const int tid  = threadIdx.x;
    const int lane = tid & 31;
    const int w    = tid >> 5;
    const int wave = blockIdx.x * 8 + w;
    const int nwaves = NEG_BLOCKS * 8;

    const float4* ne4 = (const float4*)node_embed;
    const float4* ce4 = (const float4*)ctx_embed;

    float local = 0.f;
    for (int pg = wave; pg < TOTAL_PAIRS; pg += nwaves) {
        int b  = pg / NPAIR;
        int p  = pg - b * NPAIR;
        int dl = idx_dst[p];
        int anode = batch_walk[b * WALK + dl];
        float4 a4 = ne4[(long long)anode * (EMB / 4) + lane];
        #pragma unroll
        for (int s = 0; s < 5; ++s) {
            int cpos  = neg_dst_idx[pg * 5 + s];
            int cnode = batch_walk[cpos];
            float4 c4 = ce4[(long long)cnode * (EMB / 4) + lane];
            float d = a4.x * c4.x + a4.y * c4.y + a4.z * c4.z + a4.w * c4.w;
            for (int off = 16; off; off >>= 1) d += __shfl_xor(d, off, 32);
            d = fminf(6.f, fmaxf(-6.f, d));
            local += softplus_f(d);          // -log_sigmoid(-d) == softplus(d)
        }
    }
    // `local` is identical across lanes of the wave after the butterflies.
    if (lane == 0) red[w] = local;
    __syncthreads();
    if (tid == 0) {
        float s = 0.f;
        #pragma unroll
        for (int i = 0; i < 8; ++i) s += red[i];
        partial[blockIdx.x] = (double)s;
    }
}

// ---------------------------------------------------------------------------
// Deterministic final reduction: fixed-order strided sum + tree reduce.
// ---------------------------------------------------------------------------
__global__ __launch_bounds__(256) void finalize_kernel(
    const double* __restrict__ posP,   // [BATCH]
    const double* __restrict__ negP,   // [NEG_BLOCKS]
    float* __restrict__ out)
{
    __shared__ double red[256];
    const int tid = threadIdx.x;
    double s = 0.0;
    for (int i = tid; i < BATCH; i += 256)      s += posP[i];
    for (int i = tid; i < NEG_BLOCKS; i += 256) s += negP[i];
    red[tid] = s;
    __syncthreads();
    for (int off = 128; off; off >>= 1) {
        if (tid < off) red[tid] += red[tid + off];
        __syncthreads();
    }
    if (tid == 0) out[0] = (float)(red[0] / (double)TOTAL_PAIRS);
}

extern "C" void kernel_launch(void* const* d_in, const int* in_sizes, int n_in,
                              void* d_out, int out_size, void* d_ws, size_t ws_size,
                              hipStream_t stream) {
    (void)in_sizes; (void)n_in; (void)out_size; (void)ws_size;

    const float* node_embed  = (const float*)d_in[0];
    const float* ctx_embed   = (const float*)d_in[1];
    const int*   batch_walk  = (const int*)d_in[2];
    const int*   neg_dst_idx = (const int*)d_in[3];
    float*       out         = (float*)d_out;

    char* ws = (char*)d_ws;
    int*    idx_src = (int*)ws;                              // 370 ints
    int*    idx_dst = (int*)(ws + 2048);                     // 370 ints
    double* posP    = (double*)(ws + 4096);                  // 1024 doubles
    double* negP    = (double*)(ws + 4096 + 8192);           // 2048 doubles

    build_pairs_kernel<<<1, 512, 0, stream>>>(idx_src, idx_dst);
    pos_kernel<<<BATCH, POS_THREADS, 0, stream>>>(node_embed, ctx_embed, batch_walk,
                                                  idx_src, idx_dst, posP);
    neg_kernel<<<NEG_BLOCKS, 256, 0, stream>>>(node_embed, ctx_embed, batch_walk,
                                               neg_dst_idx, idx_dst, negP);
    finalize_kernel<<<1, 256, 0, stream>>>(posP, negP, out);
}